// MultiIndexSelect_79817672228967
// MI455X (gfx1250) — compile-verified
//
#include <hip/hip_runtime.h>
#include <stdint.h>

// ---------------------------------------------------------------------------
// MultiIndexSelect on MI455X (gfx1250)
//
// out[l, idx_to[j], :] = mat[l, idx_from[j], :]   (C = 128 f32 -> 512 B rows)
//
// Pure-bandwidth kernel: ~819 MB total traffic -> ~35 us at 23.3 TB/s.
// Strategy: one wave32 moves one 512-byte row (16 B per lane, B128 ops),
// staged through LDS with the CDNA5 async copy engine so the wave never
// waits on load *data*, only on ASYNCcnt. Double-buffered batches of 4 rows
// keep 8 x 512 B transfers in flight per wave between s_wait_asynccnt fences.
// ---------------------------------------------------------------------------

#define WAVE            32
#define WAVES_PER_BLOCK 8
#define BLOCK           (WAVE * WAVES_PER_BLOCK)
#define BATCH           4              // rows per batch per wave
#define ROW_BYTES       512            // 128 f32
#define LDS_PER_WAVE    (2 * BATCH * ROW_BYTES)   // double buffer: 4 KB/wave

__global__ __launch_bounds__(BLOCK)
void MultiIndexSelect_gather_rows_async(const float* __restrict__ src,  // (L, n_src, 128)
                                        const int*   __restrict__ idx_from,
                                        const int*   __restrict__ idx_to,
                                        float*       __restrict__ dst,  // (L, out_rows, 128)
                                        int n_sel, int n_src, int out_rows)
{
    __shared__ char smem[WAVES_PER_BLOCK * LDS_PER_WAVE];   // 32 KB / block

    const int lane     = threadIdx.x & (WAVE - 1);
    const int waveIn   = threadIdx.x >> 5;
    const int laneByte = lane * 16;
    const int layer    = blockIdx.y;                  // L mapped to grid.y

    const int waveId = blockIdx.x * WAVES_PER_BLOCK + waveIn;
    const int nWaves = gridDim.x * WAVES_PER_BLOCK;
    const int stride = nWaves * BATCH;

    // Low 32 bits of a generic pointer to LDS == workgroup-relative LDS offset
    // (generic LDS addresses are {aperture_hi32, lds_offset32}).
    const uint32_t ldsBase =
        (uint32_t)(uintptr_t)(&smem[waveIn * LDS_PER_WAVE]);

    const uint64_t srcBase = (uint64_t)(uintptr_t)src +
                             (uint64_t)layer * (uint64_t)n_src * ROW_BYTES +
                             (uint64_t)laneByte;
    const uint64_t dstBase = (uint64_t)(uintptr_t)dst +
                             (uint64_t)layer * (uint64_t)out_rows * ROW_BYTES +
                             (uint64_t)laneByte;

    // Issue BATCH async row-loads (global -> LDS) for rows [b, b+BATCH).
    auto issue_loads = [&](int b, int buf) {
        const uint32_t bufBase = ldsBase + (uint32_t)(buf * (BATCH * ROW_BYTES));
        #pragma unroll
        for (int s = 0; s < BATCH; ++s) {
            int j = b + s;
            if (j < n_sel) {
                uint64_t ga  = srcBase + (uint64_t)idx_from[j] * ROW_BYTES;
                uint32_t lds = bufBase + (uint32_t)(s * ROW_BYTES) + (uint32_t)laneByte;
                asm volatile("global_load_async_to_lds_b128 %0, %1, off th:TH_LOAD_NT"
                             :: "v"(lds), "v"(ga) : "memory");
            }
        }
    };

    // Issue BATCH async row-stores (LDS -> global) for rows [b, b+BATCH).
    auto issue_stores = [&](int b, int buf) {
        const uint32_t bufBase = ldsBase + (uint32_t)(buf * (BATCH * ROW_BYTES));
        #pragma unroll
        for (int s = 0; s < BATCH; ++s) {
            int j = b + s;
            if (j < n_sel) {
                uint64_t ga  = dstBase + (uint64_t)idx_to[j] * ROW_BYTES;
                uint32_t lds = bufBase + (uint32_t)(s * ROW_BYTES) + (uint32_t)laneByte;
                asm volatile("global_store_async_from_lds_b128 %0, %1, off th:TH_STORE_NT"
                             :: "v"(ga), "v"(lds) : "memory");
            }
        }
    };

    int base = waveId * BATCH;
    if (base >= n_sel) return;

    int p = 0;
    issue_loads(base, p);                 // prologue: fill buffer 0
    for (;;) {
        int next = base + stride;
        // Fence: loads into buf p complete (and previous batch's stores
        // out of buf p^1 have drained, so buf p^1 reuse below is safe).
        asm volatile("s_wait_asynccnt 0" ::: "memory");
        if (next < n_sel) issue_loads(next, p ^ 1);   // overlap with stores
        issue_stores(base, p);
        p ^= 1;
        base = next;
        if (base >= n_sel) break;
    }
    // S_ENDPGM performs an implicit wait-idle, draining outstanding stores.
}

// ---------------------------------------------------------------------------
// Host-side launcher. Inputs (setup_inputs order):
//   0: mat1 (L, N_SRC, 128) f32     1: mat2 (L, N_SRC, 128) f32
//   2: idx_from1 (N_SEL) int        3: idx_to1 (N_SEL) int
//   4: idx_from2 (N_SEL) int        5: idx_to2 (N_SEL) int
// Output: (L, N_SEL1 + N_SEL2, 128) f32.  idx_to1 ∪ idx_to2 partitions the
// output rows, so every output element is written -> no pre-zeroing needed.
// ---------------------------------------------------------------------------
extern "C" void kernel_launch(void* const* d_in, const int* in_sizes, int n_in,
                              void* d_out, int out_size, void* d_ws, size_t ws_size,
                              hipStream_t stream)
{
    (void)n_in; (void)out_size; (void)d_ws; (void)ws_size;

    const float* mat1      = (const float*)d_in[0];
    const float* mat2      = (const float*)d_in[1];
    const int*   idx_from1 = (const int*)d_in[2];
    const int*   idx_to1   = (const int*)d_in[3];
    const int*   idx_from2 = (const int*)d_in[4];
    const int*   idx_to2   = (const int*)d_in[5];
    float*       out       = (float*)d_out;

    const int C = 128, L = 2;
    const int n_sel1   = in_sizes[2];
    const int n_sel2   = in_sizes[4];
    const int n_src    = in_sizes[0] / (L * C);
    const int out_rows = n_sel1 + n_sel2;

    auto launch = [&](const float* src, const int* ifrom, const int* ito, int n_sel) {
        if (n_sel <= 0) return;
        long waves_needed  = ((long)n_sel + BATCH - 1) / BATCH;
        long blocks_needed = (waves_needed + WAVES_PER_BLOCK - 1) / WAVES_PER_BLOCK;
        int  blocks = (int)(blocks_needed < 1024 ? blocks_needed : 1024);
        dim3 grid((unsigned)blocks, (unsigned)L);
        hipLaunchKernelGGL(MultiIndexSelect_gather_rows_async,
                           grid, dim3(BLOCK), 0, stream,
                           src, ifrom, ito, out, n_sel, n_src, out_rows);
    };

    launch(mat1, idx_from1, idx_to1, n_sel1);
    launch(mat2, idx_from2, idx_to2, n_sel2);
}